// RasterPoints_76209899700352
// MI455X (gfx1250) — compile-verified
//
#include <hip/hip_runtime.h>
#include <hip/hip_bf16.h>
#include <stdint.h>

#define SDF_H 200
#define SDF_W 200
#define BATCH 256
#define NPTS  25

// Fixed launch geometry for the zero-fill kernel so the 4x-unrolled async
// store offsets are compile-time immediates that fit IOFFSET's signed 24 bits.
#define ZB_BLOCKS   500
#define ZB_THREADS  256
#define ZB_STRIDE   (ZB_BLOCKS * ZB_THREADS)        // 128000 threads
#define ZB_STEP_B   (ZB_STRIDE * 16)                // 2,048,000 bytes / unroll step
static_assert(3LL * ZB_STEP_B < (1 << 23), "IOFFSET immediate out of range");

// ---------------------------------------------------------------------------
// Bulk zero-fill of the 1.024 GB output via the gfx1250 async LDS->global
// store engine. Each wave32 instruction moves 32 lanes x 16B = 512B from a
// 4 KB zeroed LDS staging buffer to global memory with no VGPR data operands.
// Tracked with ASYNCcnt, drained with s_wait_asynccnt. 4x unrolled using the
// 24-bit IOFFSET immediate: one 64-bit address add per 2 KB of stores.
// Pure store-bandwidth bound: ~44 us floor at 23.3 TB/s.
// ---------------------------------------------------------------------------
__global__ __launch_bounds__(ZB_THREADS) void zero_async_kernel(float* __restrict__ out,
                                                                long long n_chunks16) {
    __shared__ __align__(16) float4 zbuf[ZB_THREADS];   // 4 KB of zeros

    const int tid = threadIdx.x;
    zbuf[tid] = make_float4(0.f, 0.f, 0.f, 0.f);
    __syncthreads();  // s_wait_dscnt + barrier: LDS zeros visible to async engine

    // Per-lane LDS byte offset (wave32). Each wave reads a contiguous 512B
    // region of zeros; sequential addresses across 64 banks -> conflict-free.
    const unsigned lds_off = (unsigned)(size_t)&zbuf[tid & 31];

    const long long stride = ZB_STRIDE;  // matches launch geometry
    long long c = (long long)blockIdx.x * blockDim.x + threadIdx.x;
    const uint64_t base = (uint64_t)(uintptr_t)out;

    // Main loop: 4 async 512B wave-stores per address computation.
    for (; c + 3 * stride < n_chunks16; c += 4 * stride) {
        uint64_t ga = base + (uint64_t)c * 16u;
        asm volatile(
            "global_store_async_from_lds_b128 %0, %1, off\n\t"
            "global_store_async_from_lds_b128 %0, %1, off offset:%2\n\t"
            "global_store_async_from_lds_b128 %0, %1, off offset:%3\n\t"
            "global_store_async_from_lds_b128 %0, %1, off offset:%4"
            :: "v"(ga), "v"(lds_off),
               "i"(ZB_STEP_B), "i"(2 * ZB_STEP_B), "i"(3 * ZB_STEP_B)
            : "memory");
    }
    // Tail (empty when n_chunks16 % (4*ZB_STRIDE) == 0, as with 64M chunks).
    for (; c < n_chunks16; c += stride) {
        uint64_t ga = base + (uint64_t)c * 16u;
        asm volatile("global_store_async_from_lds_b128 %0, %1, off"
                     :: "v"(ga), "v"(lds_off)
                     : "memory");
    }
    // Drain ASYNCcnt so the stream-ordered scatter kernel sees all zeros.
    asm volatile("s_wait_asynccnt 0" ::: "memory");
}

// ---------------------------------------------------------------------------
// Scatter 1.0f at (b, row, col, p). 6400 threads total — negligible cost.
// Reproduces reference math: idx = trunc(point / resolution + origin),
// cols = idx[...,0], rows = idx[...,1]. OOB indices dropped (JAX scatter
// default); with the given inputs all indices land in [10,190).
// ---------------------------------------------------------------------------
__global__ __launch_bounds__(256) void scatter_ones_kernel(const float* __restrict__ x,
                                                           const float* __restrict__ res,
                                                           const float* __restrict__ org,
                                                           float* __restrict__ out) {
    int t = blockIdx.x * blockDim.x + threadIdx.x;
    if (t >= BATCH * NPTS) return;

    int b = t / NPTS;
    int p = t - b * NPTS;

    float px = x[b * (2 * NPTS) + 2 * p + 0];
    float py = x[b * (2 * NPTS) + 2 * p + 1];
    float rx = res[2 * b + 0];
    float ry = res[2 * b + 1];
    float ox = org[2 * b + 0];
    float oy = org[2 * b + 1];

    int col = (int)(px / rx + ox);   // trunc toward zero == astype(int32) here
    int row = (int)(py / ry + oy);

    if ((unsigned)row < (unsigned)SDF_H && (unsigned)col < (unsigned)SDF_W) {
        long long addr = ((((long long)b * SDF_H + row) * SDF_W) + col) * NPTS + p;
        out[addr] = 1.0f;
    }
}

extern "C" void kernel_launch(void* const* d_in, const int* in_sizes, int n_in,
                              void* d_out, int out_size, void* d_ws, size_t ws_size,
                              hipStream_t stream) {
    (void)in_sizes; (void)n_in; (void)d_ws; (void)ws_size;

    const float* x   = (const float*)d_in[0];  // (256, 50)
    const float* res = (const float*)d_in[1];  // (256, 2)
    const float* org = (const float*)d_in[2];  // (256, 2)
    float* out = (float*)d_out;                // (256, 200, 200, 25) = 256e6 floats

    // 256e6 floats -> 64e6 16-byte chunks; 64e6 / 128000 = exactly 500/thread.
    long long n_chunks16 = (long long)out_size / 4;

    zero_async_kernel<<<ZB_BLOCKS, ZB_THREADS, 0, stream>>>(out, n_chunks16);

    int n_scatter = BATCH * NPTS;  // 6400
    scatter_ones_kernel<<<(n_scatter + 255) / 256, 256, 0, stream>>>(x, res, org, out);
}